// Attention_69045894251074
// MI455X (gfx1250) — compile-verified
//
#include <hip/hip_runtime.h>
#include <math.h>

// Geometry (matches reference)
#define B_      16
#define H_      32
#define HKV_    8
#define G_      4
#define D_      128
#define BLOCK_  16
#define MAXB_   256
#define NBLK_   4096              // NUM_BLOCKS = B*MAX_BLOCKS
#define KSTRIDE 132               // padded LDS row stride (conflict-free)
#define NWAVES  4
#define SCALE_F 0.08838834764831845f

typedef float v2f __attribute__((ext_vector_type(2)));
typedef float v8f __attribute__((ext_vector_type(8)));

#ifndef __has_builtin
#define __has_builtin(x) 0
#endif

#if __has_builtin(__builtin_amdgcn_global_load_async_to_lds_b128)
#define HAVE_ASYNC_LDS 1
#else
#define HAVE_ASYNC_LDS 0
#endif

#if __has_builtin(__builtin_amdgcn_s_wait_asynccnt)
#define WAIT_ASYNCCNT(n) __builtin_amdgcn_s_wait_asynccnt(n)
#else
#define WAIT_ASYNCCNT(n) __asm__ volatile("s_wait_asynccnt %0" ::"i"(n) : "memory")
#endif
#define WAIT_DSCNT0() __asm__ volatile("s_wait_dscnt 0" ::: "memory")

// Exact pointer types the async-LDS builtin wants:
// param0: v4i in AS1 (global/"__device__"), param1: v4i in AS3 (LDS/"__shared__")
typedef int v4i __attribute__((vector_size(4 * sizeof(int))));
typedef __attribute__((address_space(1))) v4i gv4i_t;
typedef __attribute__((address_space(3))) v4i lv4i_t;

// ---- LDS layout (floats) ----
// qb   : [16][132] Q in B-fragment layout, rows 4..15 zero          (2112)
// per wave w: kt0/kt1 [16][132] double-buffered K tiles (2x2112)
//             pt [16][17] P tile (272)                              (4496)
// msh  : [4][16] per-wave running max                               (64)
// acc  : [128][16] combined O^T                                     (2048)
// lacc : [16] combined l                                            (16)
#define OFF_QB        0
#define OFF_KT(w, pb) (2112 + (w) * 4496 + (pb) * 2112)
#define OFF_PT(w)     (2112 + (w) * 4496 + 4224)
#define OFF_MSH       (2112 + NWAVES * 4496)
#define OFF_ACC       (OFF_MSH + 64)
#define OFF_LACC      (OFF_ACC + 2048)
#define SMEM_FLOATS   (OFF_LACC + 16)   // 22224 floats = 88896 B (< 320 KB WGP)

__global__ __launch_bounds__(NWAVES * 32)
void pa_decode_wmma(const float* __restrict__ query,
                    const float* __restrict__ key,
                    const float* __restrict__ value,
                    const float* __restrict__ kvc,
                    const int*   __restrict__ btab,
                    const int*   __restrict__ slens,
                    float*       __restrict__ out) {
  extern __shared__ float smem[];

  const int tid  = threadIdx.x;
  const int lane = tid & 31;
  const int w    = tid >> 5;      // wave id within WG
  const int h    = lane >> 4;     // lane half
  const int ml   = lane & 15;
  const int b    = blockIdx.x >> 3;
  const int kv   = blockIdx.x & 7;

  int seq = slens[b];
  if (seq < 1) seq = 1;
  const int nch    = (seq + 15) >> 4;   // chunks == vLLM cache blocks
  const int newpos = seq - 1;

  float* qb  = smem + OFF_QB;
  float* pt  = smem + OFF_PT(w);
  float* msh = smem + OFF_MSH;
  float* acc = smem + OFF_ACC;          // lacc = acc + 2048 (contiguous)

  const float* kcache = kvc;
  const float* vcache = kvc + (size_t)NBLK_ * BLOCK_ * HKV_ * D_;
  const size_t kvoff  = (size_t)kv * D_;
  const size_t newoff = (size_t)b * (HKV_ * D_) + kvoff;

  // zero combine accumulators (acc + lacc contiguous)
  for (int idx = tid; idx < 2048 + 16; idx += NWAVES * 32) acc[idx] = 0.0f;

  // stage Q: rows 0..3 = the 4 grouped query heads, rows 4..15 = 0
  for (int idx = tid; idx < 16 * 128; idx += NWAVES * 32) {
    const int r = idx >> 7, d = idx & 127;
    float qv = 0.0f;
    if (r < G_) qv = query[(size_t)b * (H_ * D_) + (size_t)(kv * G_ + r) * D_ + d];
    qb[r * KSTRIDE + d] = qv;
  }
  __syncthreads();

  // K-tile stage: one async b128 row-copy per instruction (512 B/row),
  // new-token row substituted from `key` (cache is never mutated).
  auto stageK = [&](int j, float* dst) {
    const int jblk  = btab[b * MAXB_ + j];
    const int jpos0 = j << 4;
#pragma unroll
    for (int r = 0; r < 16; ++r) {
      const float* src = (jpos0 + r == newpos)
          ? (key + newoff)
          : (kcache + (size_t)(jblk * BLOCK_ + r) * (HKV_ * D_) + kvoff);
#if HAVE_ASYNC_LDS
      __builtin_amdgcn_global_load_async_to_lds_b128(
          (gv4i_t*)(src + lane * 4),
          (lv4i_t*)(dst + r * KSTRIDE + lane * 4), 0, 0);
#else
      float4 kk = ((const float4*)src)[lane];
      *(float4*)(dst + r * KSTRIDE + lane * 4) = kk;
#endif
    }
  };

  v8f o[8];
  {
    v8f z = {};
#pragma unroll
    for (int t = 0; t < 8; ++t) o[t] = z;
  }
  float mrun = -__builtin_inff();
  float lrun = 0.0f;

  int pb = 0;
  if (w < nch) stageK(w, smem + OFF_KT(w, 0));   // prime the pipeline

  for (int i = w; i < nch; i += NWAVES) {
    float* kt = smem + OFF_KT(w, pb);
    const int blk  = btab[b * MAXB_ + i];
    const int pos0 = i << 4;
    const bool hasNext = (i + NWAVES) < nch;

    // kick off next K tile into the other buffer while we compute this one
    if (hasNext) stageK(i + NWAVES, smem + OFF_KT(w, pb ^ 1));
#if HAVE_ASYNC_LDS
    if (hasNext) WAIT_ASYNCCNT(16);   // in-order completion: tile i is done
    else         WAIT_ASYNCCNT(0);
#else
    WAIT_DSCNT0();
#endif

    // prefetch next V block (V is consumed directly from global)
    if (hasNext) {
      const int nb = btab[b * MAXB_ + i + NWAVES];
      const float* pbase = vcache + (size_t)(nb * BLOCK_ + ml) * (HKV_ * D_)
                         + kvoff + (size_t)h * 64;
#pragma unroll
      for (int t = 0; t < 2; ++t) __builtin_prefetch(pbase + t * 32, 0, 1);
    }

    // ---- scores^T tile [pos=M 16][q=N 16] = K · Q^T via 32 WMMAs ----
    v8f s = {};
#pragma unroll
    for (int db = 0; db < 32; ++db) {
      v2f a  = *(const v2f*)(kt + ml * KSTRIDE + db * 4 + 2 * h);  // A[m=pos][k=2h+v]
      v2f bb = *(const v2f*)(qb + ml * KSTRIDE + db * 4 + 2 * h);  // B[k][n=q]
      s = __builtin_amdgcn_wmma_f32_16x16x4_f32(false, a, false, bb,
                                                (short)0, s, false, false);
    }

    // ---- scale, mask, online softmax (column = lane&15) ----
    float p[8];
    float tmax = -__builtin_inff();
#pragma unroll
    for (int v = 0; v < 8; ++v) {
      const int pos = pos0 + v + 8 * h;        // C layout: row M = v + 8*h
      const float sv = (pos < seq) ? s[v] * SCALE_F : -__builtin_inff();
      p[v] = sv;
      tmax = fmaxf(tmax, sv);
    }
    tmax = fmaxf(tmax, __shfl_xor(tmax, 16, 32));
    const float mnew = fmaxf(mrun, tmax);
    const float fac  = __expf(mrun - mnew);    // first iter: exp(-inf)=0
    float tl = 0.0f;
#pragma unroll
    for (int v = 0; v < 8; ++v) { p[v] = __expf(p[v] - mnew); tl += p[v]; }
    tl += __shfl_xor(tl, 16, 32);
    lrun = lrun * fac + tl;
    mrun = mnew;
#pragma unroll
    for (int t = 0; t < 8; ++t)
#pragma unroll
      for (int v = 0; v < 8; ++v) o[t][v] *= fac;

    // P tile -> LDS for B-fragment re-read
#pragma unroll
    for (int v = 0; v < 8; ++v) pt[(v + 8 * h) * 17 + ml] = p[v];
    WAIT_DSCNT0();

    // ---- O^T += V^T · P : 8 dim-tiles x 4 pos-chunks of WMMAs ----
#pragma unroll
    for (int c = 0; c < 4; ++c) {
      const int r0 = c * 4 + 2 * h, r1 = r0 + 1;
      const float* vrow0 = (pos0 + r0 == newpos)
          ? (value + newoff)
          : (vcache + (size_t)(blk * BLOCK_ + r0) * (HKV_ * D_) + kvoff);
      const float* vrow1 = (pos0 + r1 == newpos)
          ? (value + newoff)
          : (vcache + (size_t)(blk * BLOCK_ + r1) * (HKV_ * D_) + kvoff);
      v2f b2;
      b2.x = pt[r0 * 17 + ml];
      b2.y = pt[r1 * 17 + ml];
#pragma unroll
      for (int t = 0; t < 8; ++t) {
        v2f a2;                                  // A2[m=dim][k=pos] = V^T
        a2.x = vrow0[t * 16 + ml];
        a2.y = vrow1[t * 16 + ml];
        o[t] = __builtin_amdgcn_wmma_f32_16x16x4_f32(false, a2, false, b2,
                                                     (short)0, o[t], false, false);
      }
    }
    pb ^= 1;
  }

  // ---- flash-decoding combine across the 4 waves ----
  __syncthreads();
  if (lane < 16) msh[w * 16 + ml] = mrun;
  __syncthreads();
  float mg = msh[ml];
#pragma unroll
  for (int ww = 1; ww < NWAVES; ++ww) mg = fmaxf(mg, msh[ww * 16 + ml]);
  const float fw = __expf(mrun - mg);   // wave with no chunks: exp(-inf)=0
#pragma unroll
  for (int t = 0; t < 8; ++t)
#pragma unroll
    for (int v = 0; v < 8; ++v)
      atomicAdd(&acc[(t * 16 + v + 8 * h) * 16 + ml], o[t][v] * fw);
  if (lane < 16) atomicAdd(&acc[2048 + ml], lrun * fw);   // lacc
  __syncthreads();

  // final normalize + store (only the 4 real query columns)
  for (int idx = tid; idx < G_ * D_; idx += NWAVES * 32) {
    const int g = idx >> 7, d = idx & 127;
    out[(size_t)b * (H_ * D_) + (size_t)(kv * G_ + g) * D_ + d] =
        acc[d * 16 + g] / acc[2048 + g];
  }
}

extern "C" void kernel_launch(void* const* d_in, const int* in_sizes, int n_in,
                              void* d_out, int out_size, void* d_ws, size_t ws_size,
                              hipStream_t stream) {
  (void)in_sizes; (void)n_in; (void)out_size; (void)d_ws; (void)ws_size;
  const float* query = (const float*)d_in[0];
  const float* key   = (const float*)d_in[1];
  const float* value = (const float*)d_in[2];
  const float* kvc   = (const float*)d_in[3];
  const int*   btab  = (const int*)d_in[4];
  const int*   slens = (const int*)d_in[5];
  // d_in[6] slot_mapping: implied by pos==seq_len-1 substitution (cache is
  // never mutated — inputs must stay intact across graph replays).
  float* out = (float*)d_out;
  pa_decode_wmma<<<dim3(B_ * HKV_), dim3(NWAVES * 32),
                   SMEM_FLOATS * sizeof(float), stream>>>(
      query, key, value, kvc, btab, slens, out);
}